// ResidualVectorQuantizer_44753559224654
// MI455X (gfx1250) — compile-verified
//
#include <hip/hip_runtime.h>

typedef __attribute__((ext_vector_type(2))) float v2f;
typedef __attribute__((ext_vector_type(8))) float v8f;

#define K_CB 8
#define C_CB 1024
#define D_CB 128
#define T_DIM 4096
#define N_TOK (8 * 4096)
#define ZQ_ELEMS ((size_t)N_TOK * K_CB * D_CB)   // 33,554,432
#define IDX_ELEMS ((size_t)N_TOK * K_CB)         // 262,144
#define MTILE 128
#define NWAVES 8
#define NTILES (C_CB / 16)                       // 64 code tiles
#define CB_PITCH 132                             // padded LDS pitch (floats)
#define NBLOCKS ((N_TOK / MTILE) * K_CB)         // 2048

// ---------------------------------------------------------------------------
// Kernel 1: half squared norms of every codebook vector: h[k][c] = 0.5*||c||^2
// ---------------------------------------------------------------------------
__global__ void rvq_cnorm_kernel(const float* __restrict__ cb,
                                 float* __restrict__ h) {
  int c = blockIdx.x * blockDim.x + threadIdx.x;  // 0..8191 (K_CB*C_CB)
  const float4* p = (const float4*)(cb + (size_t)c * D_CB);
  float s = 0.f;
#pragma unroll
  for (int i = 0; i < D_CB / 4; ++i) {
    float4 v = p[i];
    s += v.x * v.x + v.y * v.y + v.z * v.z + v.w * v.w;
  }
  h[c] = 0.5f * s;
}

// ---------------------------------------------------------------------------
// Kernel 2: main VQ kernel. Block = 128 tokens x 1 codebook. 8 waves x 32 ln.
// Wave: 16 tokens; A-panel (16x128 f32) resident in 64 VGPRs; B-tiles
// (16 codes x 128) double-buffered in LDS via GLOBAL_LOAD_ASYNC_TO_LDS_B128;
// scores via chained V_WMMA_F32_16X16X4_F32.
// ---------------------------------------------------------------------------
__global__ __launch_bounds__(256)
void rvq_main_kernel(const float* __restrict__ z,
                     const float* __restrict__ cb,
                     const float* __restrict__ hbuf,
                     float* __restrict__ out,
                     float* __restrict__ partial) {
  __shared__ float sC[2][16 * CB_PITCH];  // double-buffered codebook tile
  __shared__ float sH[C_CB];              // half-norms for this codebook
  __shared__ int sIdx[MTILE];             // best index per token in block
  __shared__ float sWSum[NWAVES];         // per-wave loss partials

  const int tid = threadIdx.x;
  const int wave = tid >> 5;
  const int lane = tid & 31;
  const int half = lane >> 4;   // 0: K 0..1 / rows 0..7 ; 1: K 2..3 / rows 8..15
  const int m = lane & 15;      // token-row (A) / code-col (B) within tile

  const int k = blockIdx.x & (K_CB - 1);
  const int mt = blockIdx.x >> 3;
  const int tokBase = mt * MTILE;
  const int waveTok = tokBase + wave * 16;

  const float* cbk = cb + (size_t)k * C_CB * D_CB;

  // stage half-norms: 256 threads x float4 = 1024 floats
  {
    const float4* src = (const float4*)(hbuf + k * C_CB);
    ((float4*)sH)[tid] = src[tid];
  }

  // Load resident A fragments: lane L holds x[row=waveTok+m][4s + 2*half + {0,1}]
  const float* xrow = z + (size_t)(waveTok + m) * (K_CB * D_CB) + k * D_CB;
  v2f af[32];
#pragma unroll
  for (int s = 0; s < 32; ++s) {
    float2 t = *(const float2*)(xrow + 4 * s + 2 * half);
    af[s].x = t.x;
    af[s].y = t.y;
  }

  // --- async codebook-tile staging setup ---------------------------------
  // Each thread owns 8 floats of the 16x128 tile: row=tid>>4, col=(tid&15)*8.
  // Two GLOBAL_LOAD_ASYNC_TO_LDS_B128 per thread per tile (GVS mode:
  // LDS addr from vdst, global addr saddr + vaddr; offset applies to both sides).
  // s_wait_xcnt 0 inside the asm: the compiler may immediately reuse the
  // address VGPRs (it can't see that inline asm issued VMEM), so we must
  // retire address translation (XCNT) before leaving the asm block.
  const int crow = tid >> 4;         // 0..15
  const int ccol = (tid & 15) * 8;   // 0..120
  const unsigned ldsBuf0 = (unsigned)(uintptr_t)(&sC[0][0]);
  const unsigned ldsBuf1 = (unsigned)(uintptr_t)(&sC[1][0]);
  const unsigned ldsOff = (unsigned)((crow * CB_PITCH + ccol) * 4);
  const unsigned long long cbk64 = (unsigned long long)(uintptr_t)cbk;

  // prologue: async-stage tile 0 into buffer 0
  {
    unsigned d0 = ldsBuf0 + ldsOff;
    unsigned v0 = (unsigned)((crow * D_CB + ccol) * 4);
    asm volatile(
        "global_load_async_to_lds_b128 %0, %1, %2\n\t"
        "global_load_async_to_lds_b128 %0, %1, %2 offset:16\n\t"
        "s_wait_xcnt 0x0"
        :
        : "v"(d0), "v"(v0), "s"(cbk64)
        : "memory");
  }
  asm volatile("s_wait_asynccnt 0x0" ::: "memory");
  __syncthreads();

  v8f acc = {};
  float best[8];
  int bidx[8];
#pragma unroll
  for (int i = 0; i < 8; ++i) {
    best[i] = -3.0e38f;
    bidx[i] = 0;
  }

  for (int ct = 0; ct < NTILES; ++ct) {
    // async-stage next tile; its buffer was last read in compute(ct-1),
    // which the barrier at the end of the previous iteration retired.
    if (ct + 1 < NTILES) {
      unsigned d0 = (((ct + 1) & 1) ? ldsBuf1 : ldsBuf0) + ldsOff;
      unsigned v0 = (unsigned)(((ct + 1) * 16 * D_CB + crow * D_CB + ccol) * 4);
      asm volatile(
          "global_load_async_to_lds_b128 %0, %1, %2\n\t"
          "global_load_async_to_lds_b128 %0, %1, %2 offset:16\n\t"
          "s_wait_xcnt 0x0"
          :
          : "v"(d0), "v"(v0), "s"(cbk64)
          : "memory");
    }

    // 16x16 score tile, K accumulated over d=128 in 32 chained f32 WMMAs
    const float* brow = &sC[ct & 1][m * CB_PITCH + 2 * half];
#pragma unroll
    for (int s = 0; s < 32; ++s) {
      float2 t = *(const float2*)(brow + 4 * s);
      v2f bv;
      bv.x = t.x;
      bv.y = t.y;
      acc = __builtin_amdgcn_wmma_f32_16x16x4_f32(false, af[s], false, bv,
                                                  (short)0, acc, false, false);
    }

    // score = dot - 0.5*||c||^2 ; running argmax (ties -> lowest index)
    const float hl = sH[ct * 16 + m];
    const int code = ct * 16 + m;
#pragma unroll
    for (int i = 0; i < 8; ++i) {
      float v = acc[i] - hl;
      if (v > best[i]) {
        best[i] = v;
        bidx[i] = code;
      }
      acc[i] = 0.0f;
    }

    // my async writes for tile ct+1 are done; one barrier publishes them
    // and retires everyone's reads of buffer ct for the next iteration.
    asm volatile("s_wait_asynccnt 0x0" ::: "memory");
    __syncthreads();
  }

  // cross-lane reduction within each 16-lane half (rows i / i+8 per VGPR i)
#pragma unroll
  for (int i = 0; i < 8; ++i) {
    float bv = best[i];
    int bi = bidx[i];
#pragma unroll
    for (int off = 8; off >= 1; off >>= 1) {
      float ov = __shfl_xor(bv, off, 32);
      int oi = __shfl_xor(bi, off, 32);
      if (ov > bv || (ov == bv && oi < bi)) {
        bv = ov;
        bi = oi;
      }
    }
    best[i] = bv;
    bidx[i] = bi;
  }

  // lanes 0 and 16 publish the 16 winners; also write indices output (B,K,T)
  if (m == 0) {
#pragma unroll
    for (int i = 0; i < 8; ++i) {
      int tokLocal = wave * 16 + i + 8 * half;
      sIdx[tokLocal] = bidx[i];
      int n = tokBase + tokLocal;
      int b = n / T_DIM;
      int t = n - b * T_DIM;
      out[ZQ_ELEMS + (size_t)b * K_CB * T_DIM + (size_t)k * T_DIM + t] =
          (float)bidx[i];
    }
  }
  __syncthreads();

  // gather q, write z_q (coalesced 512B per row), accumulate loss
  float lsum = 0.f;
  for (int r = 0; r < 16; ++r) {
    int tokLocal = wave * 16 + r;
    int n = tokBase + tokLocal;
    int idx = sIdx[tokLocal];
    float4 q = *(const float4*)(cbk + (size_t)idx * D_CB + lane * 4);
    float4 xv = *(const float4*)(z + (size_t)n * (K_CB * D_CB) + k * D_CB + lane * 4);
    *(float4*)(out + (size_t)n * (K_CB * D_CB) + k * D_CB + lane * 4) = q;
    float dx = q.x - xv.x, dy = q.y - xv.y, dzv = q.z - xv.z, dw = q.w - xv.w;
    lsum += dx * dx + dy * dy + dzv * dzv + dw * dw;
  }
#pragma unroll
  for (int off = 16; off >= 1; off >>= 1) lsum += __shfl_xor(lsum, off, 32);
  if (lane == 0) sWSum[wave] = lsum;
  __syncthreads();
  if (tid == 0) {
    float s = 0.f;
#pragma unroll
    for (int i = 0; i < NWAVES; ++i) s += sWSum[i];  // fixed order: deterministic
    partial[blockIdx.x] = s;
  }
}

// ---------------------------------------------------------------------------
// Kernel 3: deterministic tree-reduce of 2048 block partials -> vq_loss
// ---------------------------------------------------------------------------
__global__ void rvq_loss_reduce_kernel(const float* __restrict__ partial,
                                       float* __restrict__ out) {
  __shared__ float sm[256];
  float s = 0.f;
  for (int i = threadIdx.x; i < NBLOCKS; i += 256) s += partial[i];
  sm[threadIdx.x] = s;
  __syncthreads();
  for (int st = 128; st >= 1; st >>= 1) {
    if ((int)threadIdx.x < st) sm[threadIdx.x] += sm[threadIdx.x + st];
    __syncthreads();
  }
  if (threadIdx.x == 0) {
    out[ZQ_ELEMS + IDX_ELEMS] = sm[0] * (1.0f / ((float)N_TOK * (float)D_CB));
  }
}

// ---------------------------------------------------------------------------
extern "C" void kernel_launch(void* const* d_in, const int* in_sizes, int n_in,
                              void* d_out, int out_size, void* d_ws,
                              size_t ws_size, hipStream_t stream) {
  (void)in_sizes;
  (void)n_in;
  (void)out_size;
  (void)ws_size;
  const float* z = (const float*)d_in[0];
  const float* cb = (const float*)d_in[1];
  float* out = (float*)d_out;
  float* h = (float*)d_ws;          // 8192 floats
  float* partial = h + K_CB * C_CB; // 2048 floats

  rvq_cnorm_kernel<<<(K_CB * C_CB) / 256, 256, 0, stream>>>(cb, h);
  rvq_main_kernel<<<NBLOCKS, 256, 0, stream>>>(z, cb, h, out, partial);
  rvq_loss_reduce_kernel<<<1, 256, 0, stream>>>(partial, out);
}